// SparseGNNLayer_16630113370839
// MI455X (gfx1250) — compile-verified
//
#include <hip/hip_runtime.h>

#define N_NODES 50000
#define N_EDGES 500000
#define D_H 128
#define D_E 32
#define M_OUT 128
#define U_OUT 128

// LDS k-strides for transposed weight tiles [n][k].
// 166: 166%64=38, gcd(38,64)=2 -> lanes 0..31 hit distinct banks; 166*4%8==0 -> rows 8B aligned.
// 262: 262%64=6,  gcd(6,64)=2  -> conflict-free; rows 8B aligned.
#define WM_STRIDE 166
#define WU_STRIDE 262

typedef __attribute__((ext_vector_type(2))) float v2f;
typedef __attribute__((ext_vector_type(8))) float v8f;

__device__ __forceinline__ v8f wmma_f32(v2f a, v2f b, v8f c) {
  // D = A(16x4 f32) * B(4x16 f32) + C(16x16 f32)
  return __builtin_amdgcn_wmma_f32_16x16x4_f32(false, a, false, b, (short)0, c,
                                               false, false);
}

__device__ __forceinline__ void atomic_add_f32(float* p, float v) {
  unsafeAtomicAdd(p, v);  // lowers to global_atomic_add_f32 (no return)
}

__global__ void zero_f4_kernel(float4* __restrict__ p, int n4) {
  int i = blockIdx.x * blockDim.x + threadIdx.x;
  if (i < n4) p[i] = make_float4(0.f, 0.f, 0.f, 0.f);
}

// ---------------------------------------------------------------------------
// Edge message kernel: Y = ReLU([H[src]|Xe] @ W_M + b_M); atomic scatter to Z.
// One wave per 16-edge tile; 8 v8f accumulators = full 16x128 output tile.
// ---------------------------------------------------------------------------
__global__ __launch_bounds__(256) void gnn_edge_kernel(
    const float* __restrict__ H, const float* __restrict__ Xe,
    const int* __restrict__ idx, const float* __restrict__ W_M,
    const float* __restrict__ b_M, float* __restrict__ Z) {
  extern __shared__ float smem[];
  float* Wt = smem;                         // [M_OUT][WM_STRIDE] = W_M^T
  float* bs = smem + M_OUT * WM_STRIDE;     // [M_OUT]

  for (int i = threadIdx.x; i < (D_H + D_E) * M_OUT; i += blockDim.x) {
    int k = i >> 7, n = i & 127;            // W_M is [160][128] row-major
    Wt[n * WM_STRIDE + k] = W_M[i];
  }
  if (threadIdx.x < M_OUT) bs[threadIdx.x] = b_M[threadIdx.x];
  __syncthreads();

  const int wave = threadIdx.x >> 5;
  const int lane = threadIdx.x & 31;
  const int l16  = lane & 15;
  const int hi   = lane >> 4;   // half-wave: lanes 16..31 carry K+2..K+3
  const int koff = hi << 1;
  const int* __restrict__ dstIdx = idx + N_EDGES;

  const int nTiles = N_EDGES / 16;
  for (int tile = blockIdx.x * 8 + wave; tile < nTiles; tile += gridDim.x * 8) {
    const int tBase = tile << 4;
    const int e = tBase + l16;
    const int s = idx[e];                           // src node
    const float* hp = H + (size_t)s * D_H + koff;
    const float* xp = Xe + (size_t)e * D_E + koff;

    v8f acc[8] = {};

    // K = 0..127 from gathered H[src]
    #pragma unroll 2
    for (int k = 0; k < D_H; k += 4) {
      v2f a = *(const v2f*)(hp + k);
      #pragma unroll
      for (int j = 0; j < 8; ++j) {
        v2f b = *(const v2f*)(&Wt[(j * 16 + l16) * WM_STRIDE + k + koff]);
        acc[j] = wmma_f32(a, b, acc[j]);
      }
    }
    // K = 128..159 from Xe
    #pragma unroll
    for (int k = 0; k < D_E; k += 4) {
      v2f a = *(const v2f*)(xp + k);
      #pragma unroll
      for (int j = 0; j < 8; ++j) {
        v2f b = *(const v2f*)(&Wt[(j * 16 + l16) * WM_STRIDE + (D_H + k) + koff]);
        acc[j] = wmma_f32(a, b, acc[j]);
      }
    }

    // Epilogue: C/D layout -> lane holds (m = r + 8*hi, n = j*16 + l16)
    #pragma unroll
    for (int r = 0; r < 8; ++r) {
      const int m = r + (hi << 3);
      const int d = dstIdx[tBase + m];
      float* zrow = Z + (size_t)d * M_OUT;
      #pragma unroll
      for (int j = 0; j < 8; ++j) {
        const int n = j * 16 + l16;
        float v = acc[j][r] + bs[n];
        v = v > 0.f ? v : 0.f;
        atomic_add_f32(zrow + n, v);
      }
    }
  }
}

// ---------------------------------------------------------------------------
// Update kernel: H_next = ReLU([H|Z] @ W_U + b_U). One wave per 16-node tile.
// ---------------------------------------------------------------------------
__global__ __launch_bounds__(256) void gnn_update_kernel(
    const float* __restrict__ H, const float* __restrict__ Z,
    const float* __restrict__ W_U, const float* __restrict__ b_U,
    float* __restrict__ out) {
  extern __shared__ float smem[];
  float* Wt = smem;                         // [U_OUT][WU_STRIDE] = W_U^T
  float* bs = smem + U_OUT * WU_STRIDE;     // [U_OUT]

  for (int i = threadIdx.x; i < (D_H + M_OUT) * U_OUT; i += blockDim.x) {
    int k = i >> 7, n = i & 127;            // W_U is [256][128] row-major
    Wt[n * WU_STRIDE + k] = W_U[i];
  }
  if (threadIdx.x < U_OUT) bs[threadIdx.x] = b_U[threadIdx.x];
  __syncthreads();

  const int wave = threadIdx.x >> 5;
  const int lane = threadIdx.x & 31;
  const int l16  = lane & 15;
  const int hi   = lane >> 4;
  const int koff = hi << 1;

  const int nTiles = N_NODES / 16;
  for (int tile = blockIdx.x * 8 + wave; tile < nTiles; tile += gridDim.x * 8) {
    const int tBase = tile << 4;
    const int node = tBase + l16;
    const float* hp = H + (size_t)node * D_H + koff;
    const float* zp = Z + (size_t)node * M_OUT + koff;

    v8f acc[8] = {};

    #pragma unroll 2
    for (int k = 0; k < D_H; k += 4) {
      v2f a = *(const v2f*)(hp + k);
      #pragma unroll
      for (int j = 0; j < 8; ++j) {
        v2f b = *(const v2f*)(&Wt[(j * 16 + l16) * WU_STRIDE + k + koff]);
        acc[j] = wmma_f32(a, b, acc[j]);
      }
    }
    #pragma unroll 2
    for (int k = 0; k < M_OUT; k += 4) {
      v2f a = *(const v2f*)(zp + k);
      #pragma unroll
      for (int j = 0; j < 8; ++j) {
        v2f b = *(const v2f*)(&Wt[(j * 16 + l16) * WU_STRIDE + (D_H + k) + koff]);
        acc[j] = wmma_f32(a, b, acc[j]);
      }
    }

    #pragma unroll
    for (int r = 0; r < 8; ++r) {
      const int m = r + (hi << 3);
      float* orow = out + (size_t)(tBase + m) * U_OUT;
      #pragma unroll
      for (int j = 0; j < 8; ++j) {
        const int n = j * 16 + l16;
        float v = acc[j][r] + bs[n];
        orow[n] = v > 0.f ? v : 0.f;
      }
    }
  }
}

extern "C" void kernel_launch(void* const* d_in, const int* in_sizes, int n_in,
                              void* d_out, int out_size, void* d_ws,
                              size_t ws_size, hipStream_t stream) {
  const float* H   = (const float*)d_in[0];
  const float* Xe  = (const float*)d_in[1];
  const int*   idx = (const int*)d_in[2];   // [2][N_EDGES]: src then dst
  const float* W_M = (const float*)d_in[3];
  const float* b_M = (const float*)d_in[4];
  const float* W_U = (const float*)d_in[5];
  const float* b_U = (const float*)d_in[6];
  float* out = (float*)d_out;
  float* Z   = (float*)d_ws;                // [N_NODES][M_OUT] accumulator

  // 1) zero Z every call (determinism: we accumulate with atomics)
  const int n4 = (N_NODES * M_OUT) / 4;
  zero_f4_kernel<<<(n4 + 255) / 256, 256, 0, stream>>>((float4*)Z, n4);

  // 2) edge messages + scatter-add (persistent grid, 8 waves/block)
  const size_t shEdge = (size_t)(M_OUT * WM_STRIDE + M_OUT) * sizeof(float);
  gnn_edge_kernel<<<1024, 256, shEdge, stream>>>(H, Xe, idx, W_M, b_M, Z);

  // 3) node update MLP
  const size_t shUpd = (size_t)(U_OUT * WU_STRIDE + U_OUT) * sizeof(float);
  const int updBlocks = (N_NODES / 16 + 7) / 8;  // 391
  gnn_update_kernel<<<updBlocks, 256, shUpd, stream>>>(H, Z, W_U, b_U, out);
}